// EngramConv_5179730559120
// MI455X (gfx1250) — compile-verified
//
#include <hip/hip_runtime.h>
#include <cstdint>

// EngramConv fused kernel for MI455X (gfx1250, wave32).
// out[b,s,d] = silu( nw[d] * sum_k w[d,k] * rinv[b, s-9+3k] * x[b, s-9+3k, d] ) + x[b,s,d]
// rinv[b,s] = rsqrt(mean_d(x^2) + 1e-6), taps before s=0 are zero (causal pad of normed x).

namespace {
constexpr int Bn   = 4;
constexpr int Sn   = 4096;
constexpr int Dn   = 2048;
constexpr int Kn   = 4;
constexpr int DIL  = 3;
constexpr int PADL = (Kn - 1) * DIL;   // 9
constexpr int WG   = 256;              // 8 wave32 waves
constexpr int NWAVE = WG / 32;
constexpr float EPSF = 1e-6f;
}

typedef unsigned int       u32;
typedef unsigned long long u64;
typedef __attribute__((ext_vector_type(4))) u32 u32x4_t;
typedef __attribute__((ext_vector_type(8))) int i32x8_t;
typedef __attribute__((ext_vector_type(4))) int i32x4_t;

__global__ __launch_bounds__(WG) void engram_fused_kernel(
    const float* __restrict__ x,    // [B,S,D]
    const float* __restrict__ nw,   // [D]
    const float* __restrict__ cw,   // [D,1,K] -> [D][4] contiguous
    float* __restrict__ out) {      // [B,S,D]
  __shared__ float rows[Kn][Dn];        // 4 raw tap rows, 32 KB
  __shared__ float redk[Kn][NWAVE];
  __shared__ float rinv_sh[Kn];

  const int tid = threadIdx.x;
  const int blk = blockIdx.x;
  const int b = blk >> 12;              // Sn == 4096
  const int s = blk & (Sn - 1);
  const size_t row_base = (((size_t)b << 12) + (size_t)s) * (size_t)Dn;

  bool staged = false;
#if __has_builtin(__builtin_amdgcn_tensor_load_to_lds) && __has_builtin(__builtin_amdgcn_s_wait_tensorcnt)
  if (s >= PADL) {
    staged = true;
    if (tid == 0) {
      // TDM descriptor: 2D tile of 4 rows x 2048 f32, rows DIL apart in x.
      const u64 ga = (u64)(uintptr_t)(x + (row_base - (size_t)PADL * Dn));
      const u32 lds_base = (u32)(uintptr_t)(&rows[0][0]);   // low 32 bits = LDS offset
      u32x4_t g0;
      g0[0] = 1u;                                   // count=1 (valid), user mode
      g0[1] = lds_base;                             // lds_addr
      g0[2] = (u32)(ga & 0xFFFFFFFFull);            // global_addr[31:0]
      g0[3] = (u32)((ga >> 32) & 0x01FFFFFFull)     // global_addr[56:32]
            | (2u << 30);                           // type = 2 ("image")
      i32x8_t g1;
      g1[0] = (int)(2u << 16);                      // workgroup_mask=0, data_size=2 (4B)
      g1[1] = (int)(((u32)Dn & 0xFFFFu) << 16);     // tensor_dim0[15:0] in [31:16]
      g1[2] = (int)(((u32)Dn >> 16)                 // tensor_dim0[31:16]
            | (((u32)Kn & 0xFFFFu) << 16));         // tensor_dim1[15:0] = 4
      g1[3] = (int)((((u32)Kn >> 16) & 0xFFFFu)     // tensor_dim1[31:16]
            | (((u32)Dn & 0xFFFFu) << 16));         // tile_dim0 = 2048
      g1[4] = (int)((u32)Kn & 0xFFFFu);             // tile_dim1 = 4, tile_dim2 = 0
      g1[5] = (int)(u32)(DIL * Dn);                 // tensor_dim0_stride = 3*D elems
      g1[6] = 0;                                    // stride hi, dim1_stride lo
      g1[7] = 0;                                    // dim1_stride hi
      i32x4_t gz4 = {0, 0, 0, 0};                   // groups 2/3 unused (2D tensor)
      i32x8_t gz8 = {0, 0, 0, 0, 0, 0, 0, 0};       // trailing group (gather/iterate): unused
      __builtin_amdgcn_tensor_load_to_lds(g0, g1, gz4, gz4, gz8, 0);
      __builtin_amdgcn_s_wait_tensorcnt(0);         // TENSORcnt of issuing wave
    }
  }
#endif
  if (!staged) {
    // Edge rows (s < 9): cooperative fill with causal zero padding.
    #pragma unroll
    for (int k = 0; k < Kn; ++k) {
      const int r = s - PADL + k * DIL;
      float4* dst = (float4*)(&rows[k][0]);
      if (r >= 0) {
        const float4* src = (const float4*)(x + (((size_t)b << 12) + (size_t)r) * (size_t)Dn);
        for (int i = tid; i < Dn / 4; i += WG) dst[i] = src[i];
      } else {
        const float4 z = make_float4(0.f, 0.f, 0.f, 0.f);
        for (int i = tid; i < Dn / 4; i += WG) dst[i] = z;
      }
    }
  }
  __syncthreads();

  // --- RMS of each of the 4 staged rows (from LDS) ---
  const int c0 = tid * 4;          // channels [c0, c0+3] and [c0+1024, c0+1027]
  const int c1 = c0 + (Dn / 2);
  float ss[Kn];
  #pragma unroll
  for (int k = 0; k < Kn; ++k) {
    float4 a = *(const float4*)(&rows[k][c0]);
    float4 c = *(const float4*)(&rows[k][c1]);
    ss[k] = a.x * a.x + a.y * a.y + a.z * a.z + a.w * a.w
          + c.x * c.x + c.y * c.y + c.z * c.z + c.w * c.w;
  }
  #pragma unroll
  for (int off = 16; off > 0; off >>= 1) {
    #pragma unroll
    for (int k = 0; k < Kn; ++k) ss[k] += __shfl_xor(ss[k], off, 32);
  }
  const int lane = tid & 31, wid = tid >> 5;
  if (lane == 0) {
    #pragma unroll
    for (int k = 0; k < Kn; ++k) redk[k][wid] = ss[k];
  }
  __syncthreads();
  if (tid < Kn) {
    float t = 0.f;
    #pragma unroll
    for (int w = 0; w < NWAVE; ++w) t += redk[tid][w];
    rinv_sh[tid] = rsqrtf(t * (1.0f / (float)Dn) + EPSF);
  }
  __syncthreads();
  const float r0 = rinv_sh[0], r1 = rinv_sh[1], r2 = rinv_sh[2], r3 = rinv_sh[3];

  // --- depthwise conv + silu + residual; 8 channels per thread ---
  const float4* wv = (const float4*)cw;   // wv[d] = {w[d][0..3]}
  const float4* nv = (const float4*)nw;
  #pragma unroll
  for (int chunk = 0; chunk < 2; ++chunk) {
    const int d = c0 + chunk * (Dn / 2);
    float4 x0 = *(const float4*)(&rows[0][d]);
    float4 x1 = *(const float4*)(&rows[1][d]);
    float4 x2 = *(const float4*)(&rows[2][d]);
    float4 x3 = *(const float4*)(&rows[3][d]);   // == raw x[b,s,d..d+3] (residual)
    float4 w0 = wv[d + 0], w1 = wv[d + 1], w2 = wv[d + 2], w3 = wv[d + 3];
    float4 g = nv[d >> 2];
    float4 y;
    y.x = g.x * (w0.x * x0.x * r0 + w0.y * x1.x * r1 + w0.z * x2.x * r2 + w0.w * x3.x * r3);
    y.y = g.y * (w1.x * x0.y * r0 + w1.y * x1.y * r1 + w1.z * x2.y * r2 + w1.w * x3.y * r3);
    y.z = g.z * (w2.x * x0.z * r0 + w2.y * x1.z * r1 + w2.z * x2.z * r2 + w2.w * x3.z * r3);
    y.w = g.w * (w3.x * x0.w * r0 + w3.y * x1.w * r1 + w3.z * x2.w * r2 + w3.w * x3.w * r3);
    float4 o;
    o.x = y.x / (1.f + expf(-y.x)) + x3.x;
    o.y = y.y / (1.f + expf(-y.y)) + x3.y;
    o.z = y.z / (1.f + expf(-y.z)) + x3.z;
    o.w = y.w / (1.f + expf(-y.w)) + x3.w;
    *(float4*)(&out[row_base + (size_t)d]) = o;
  }
}

extern "C" void kernel_launch(void* const* d_in, const int* in_sizes, int n_in,
                              void* d_out, int out_size, void* d_ws, size_t ws_size,
                              hipStream_t stream) {
  (void)in_sizes; (void)n_in; (void)out_size; (void)d_ws; (void)ws_size;
  const float* x  = (const float*)d_in[0];
  const float* nw = (const float*)d_in[1];
  const float* cw = (const float*)d_in[2];
  float* out = (float*)d_out;
  engram_fused_kernel<<<Bn * Sn, WG, 0, stream>>>(x, nw, cw, out);
}